// ConOA_60748017434937
// MI455X (gfx1250) — compile-verified
//
#include <hip/hip_runtime.h>
#include <math.h>

typedef __attribute__((ext_vector_type(2))) float v2f;
typedef __attribute__((ext_vector_type(8))) float v8f;

#define EDIM 128
#define BROWS 1024
#define QCOLS 65536
#define NORGS 2048
#define N2COLS 4096 /* bao(1024) + bpo(1024) + qoe(2048) */
#define CSPLIT 8    /* column partitions for the loss kernels */
#define NRT (BROWS / 16)              /* 64 row tiles */
#define PART_PLANE (CSPLIT * NRT * 16) /* 8192 */
#define PART_SIZE (3 * PART_PLANE)     /* m,s,p planes per loss */

/* workspace layout (float offsets) */
#define WS_ANORM 0
#define WS_ASSETSN (WS_ANORM + BROWS * EDIM)
#define WS_INVQ (WS_ASSETSN + BROWS * EDIM)
#define WS_GSUM (WS_INVQ + QCOLS)
#define WS_GCNT (WS_GSUM + NORGS * EDIM)
#define WS_CNTB (WS_GCNT + NORGS)
#define WS_SUMA (WS_CNTB + NORGS)
#define WS_SUMS (WS_SUMA + EDIM)
#define WS_BAO (WS_SUMS + EDIM)
#define WS_COLMAT2 (WS_BAO + BROWS * EDIM)
#define WS_COLIDX2 (WS_COLMAT2 + EDIM * N2COLS)
#define WS_PART (WS_COLIDX2 + N2COLS)
#define WS_ZERO_BEGIN WS_GSUM
#define WS_ZERO_COUNT (NORGS * EDIM + NORGS + NORGS + EDIM + EDIM)

__global__ __launch_bounds__(256) void k_zero(float* zreg, int n, float* out) {
  int i = blockIdx.x * 256 + threadIdx.x;
  if (i < n) zreg[i] = 0.0f;
  if (i < 3) out[i] = 0.0f;
}

/* per-row normalize anchors & assets; accumulate column sums & batch histogram */
__global__ __launch_bounds__(128) void k_rows(
    const float* __restrict__ anch, const float* __restrict__ anchm,
    const float* __restrict__ assm, const int* __restrict__ bidx,
    float* __restrict__ anorm, float* __restrict__ assetsn,
    float* __restrict__ suma, float* __restrict__ sums, float* __restrict__ cntb) {
  __shared__ float red[128];
  int i = blockIdx.x, e = threadIdx.x;
  float va = anch[i * EDIM + e];
  red[e] = va * va;
  __syncthreads();
  for (int s = 64; s > 0; s >>= 1) { if (e < s) red[e] += red[e + s]; __syncthreads(); }
  float inva = 1.0f / fmaxf(sqrtf(red[0]), 1e-12f);
  __syncthreads();
  anorm[i * EDIM + e] = va * inva;
  float vs = assm[i * EDIM + e];
  red[e] = vs * vs;
  __syncthreads();
  for (int s = 64; s > 0; s >>= 1) { if (e < s) red[e] += red[e + s]; __syncthreads(); }
  float invs = 1.0f / fmaxf(sqrtf(red[0]), 1e-12f);
  assetsn[i * EDIM + e] = vs * invs;
  atomicAdd(&suma[e], anchm[i * EDIM + e]);
  atomicAdd(&sums[e], vs);
  if (e == 0) atomicAdd(&cntb[bidx[i]], 1.0f);
}

/* per queue column: inverse norm + segment sums + segment counts */
__global__ __launch_bounds__(256) void k_queue(
    const float* __restrict__ queue, const int* __restrict__ qidx,
    float* __restrict__ gsum, float* __restrict__ gcnt, float* __restrict__ invq) {
  int j = blockIdx.x * 256 + threadIdx.x;
  int o = qidx[j];
  float ss = 0.0f;
  for (int e = 0; e < EDIM; ++e) {
    float v = queue[(size_t)e * QCOLS + j];
    ss += v * v;
    atomicAdd(&gsum[o * EDIM + e], v);
  }
  invq[j] = 1.0f / fmaxf(sqrtf(ss), 1e-12f);
  atomicAdd(&gcnt[o], 1.0f);
}

/* queue_org_emb = normalize(gsum) rows -> columns [2048..4095] of colmat2 */
__global__ __launch_bounds__(128) void k_qoe(
    const float* __restrict__ gsum, float* __restrict__ colmat2, int* __restrict__ colidx2) {
  __shared__ float red[128];
  int o = blockIdx.x, e = threadIdx.x;
  float g = gsum[o * EDIM + e];
  red[e] = g * g;
  __syncthreads();
  for (int s = 64; s > 0; s >>= 1) { if (e < s) red[e] += red[e + s]; __syncthreads(); }
  float inv = 1.0f / fmaxf(sqrtf(red[0]), 1e-12f);
  colmat2[e * N2COLS + 2 * BROWS + o] = g * inv;
  if (e == 0) colidx2[2 * BROWS + o] = o;
}

/* bao/bpo = normalize(sum + gsum[org]) -> bao rows + columns [0..2047] of colmat2 */
__global__ __launch_bounds__(128) void k_orgemb(
    const float* __restrict__ gsum, const float* __restrict__ suma,
    const float* __restrict__ sums, const int* __restrict__ bidx,
    float* __restrict__ bao, float* __restrict__ colmat2, int* __restrict__ colidx2) {
  __shared__ float red[128];
  int i = blockIdx.x, e = threadIdx.x;
  int o = bidx[i];
  float g = gsum[o * EDIM + e];
  float va = suma[e] + g;
  float vp = sums[e] + g;
  red[e] = va * va;
  __syncthreads();
  for (int s = 64; s > 0; s >>= 1) { if (e < s) red[e] += red[e + s]; __syncthreads(); }
  float inva = 1.0f / fmaxf(sqrtf(red[0]), 1e-12f);
  __syncthreads();
  red[e] = vp * vp;
  __syncthreads();
  for (int s = 64; s > 0; s >>= 1) { if (e < s) red[e] += red[e + s]; __syncthreads(); }
  float invp = 1.0f / fmaxf(sqrtf(red[0]), 1e-12f);
  float ba = va * inva;
  bao[i * EDIM + e] = ba;
  colmat2[e * N2COLS + i] = ba;
  colmat2[e * N2COLS + BROWS + i] = vp * invp;
  if (e == 0) { colidx2[i] = o; colidx2[BROWS + i] = o; }
}

/*
 * Streaming contrastive-loss GEMM (WMMA f32 16x16x4, wave32).
 * grid = (64 row tiles, CSPLIT column partitions); 8 waves per block.
 * seg1: N1 columns stored as rows ([N1][E], unit columns).
 * seg2: N2 columns K-major ([E][LD2]), optional per-column scale.
 * Writes per (partition,rowtile,row): online-softmax partial (m, s, possum).
 */
template <int N1, int N2, int LD2, bool HAS_SCALE>
__global__ __launch_bounds__(256) void k_loss(
    const float* __restrict__ A, const int* __restrict__ rowidx,
    const float* __restrict__ seg1, const int* __restrict__ s1org,
    const float* __restrict__ seg2,
    const float* __restrict__ scale2, const int* __restrict__ s2org,
    float* __restrict__ part) {
  const int lane = threadIdx.x & 31;
  const int wave = threadIdx.x >> 5;
  const int m = lane & 15; /* A row within tile / B-C-D column within tile */
  const int h = lane >> 4; /* lane half */
  const int rt = blockIdx.x;
  const int pid = blockIdx.y;
  const int R0 = rt * 16;
  const float invT = 1.0f / 0.07f;

  /* A fragments: lane m holds A[m][4k+2h], A[m][4k+2h+1] per ISA layout */
  v2f afr[32];
#pragma unroll
  for (int ks = 0; ks < 32; ++ks)
    afr[ks] = *(const v2f*)(A + (R0 + m) * EDIM + ks * 4 + 2 * h);

  /* C/D rows owned by this lane: M = r + 8*h */
  int rorg[8];
#pragma unroll
  for (int r = 0; r < 8; ++r) rorg[r] = rowidx[R0 + r + 8 * h];

  float mrun[8], srun[8], prun[8];
#pragma unroll
  for (int r = 0; r < 8; ++r) { mrun[r] = -1e30f; srun[r] = 0.0f; prun[r] = 0.0f; }

  constexpr int NTILES = (N1 + N2) >> 4;
  constexpr int NSTREAM = CSPLIT * 8;
  for (int ct = pid * 8 + wave; ct < NTILES; ct += NSTREAM) {
    const int base = ct << 4; /* wave-uniform: EXEC stays all-ones around WMMA */
    v8f acc = {0.f, 0.f, 0.f, 0.f, 0.f, 0.f, 0.f, 0.f};
    float scale;
    int corg;
    if (N1 > 0 && base < N1) {
      const float* cp = seg1 + (size_t)(base + m) * EDIM + 2 * h;
#pragma unroll
      for (int ks = 0; ks < 32; ++ks) {
        v2f bf = *(const v2f*)(cp + ks * 4); /* B[2h][n], B[2h+1][n] */
        acc = __builtin_amdgcn_wmma_f32_16x16x4_f32(false, afr[ks], false, bf,
                                                    (short)0, acc, false, false);
      }
      scale = invT;
      corg = s1org[base + m];
    } else {
      const int b2 = base - N1;
      const float* cp = seg2 + (size_t)(2 * h) * LD2 + b2 + m;
#pragma unroll
      for (int ks = 0; ks < 32; ++ks) {
        v2f bf;
        bf.x = cp[(size_t)(ks * 4) * LD2];
        bf.y = cp[(size_t)(ks * 4 + 1) * LD2];
        acc = __builtin_amdgcn_wmma_f32_16x16x4_f32(false, afr[ks], false, bf,
                                                    (short)0, acc, false, false);
      }
      scale = (HAS_SCALE ? scale2[b2 + m] : 1.0f) * invT;
      corg = s2org[b2 + m];
    }
    /* online softmax + positive-sum epilogue; this lane's column is base+m */
#pragma unroll
    for (int r = 0; r < 8; ++r) {
      float y = acc[r] * scale;
      float mo = mrun[r];
      float mn = fmaxf(mo, y);
      srun[r] = srun[r] * __expf(mo - mn) + __expf(y - mn);
      mrun[r] = mn;
      prun[r] += (corg == rorg[r]) ? y : 0.0f;
    }
  }

  /* merge the 16 column-slots within each lane half */
#pragma unroll
  for (int mask = 1; mask < 16; mask <<= 1) {
#pragma unroll
    for (int r = 0; r < 8; ++r) {
      float mo = __shfl_xor(mrun[r], mask, 32);
      float so = __shfl_xor(srun[r], mask, 32);
      float po = __shfl_xor(prun[r], mask, 32);
      float mn = fmaxf(mrun[r], mo);
      srun[r] = srun[r] * __expf(mrun[r] - mn) + so * __expf(mo - mn);
      mrun[r] = mn;
      prun[r] += po;
    }
  }

  __shared__ float lm[8][16], lsv[8][16], lp[8][16];
  if (m == 0) { /* lanes 0 (rows 0-7) and 16 (rows 8-15) */
#pragma unroll
    for (int r = 0; r < 8; ++r) {
      int row = r + 8 * h;
      lm[wave][row] = mrun[r];
      lsv[wave][row] = srun[r];
      lp[wave][row] = prun[r];
    }
  }
  __syncthreads();
  if (threadIdx.x < 16) {
    int row = threadIdx.x;
    float M = -1e30f, S = 0.0f, P = 0.0f;
    for (int w = 0; w < 8; ++w) {
      float mw = lm[w][row];
      float mn = fmaxf(M, mw);
      S = S * __expf(M - mn) + lsv[w][row] * __expf(mw - mn);
      M = mn;
      P += lp[w][row];
    }
    int idx = (pid * NRT + rt) * 16 + row;
    part[idx] = M;
    part[PART_PLANE + idx] = S;
    part[2 * PART_PLANE + idx] = P;
  }
}

/* merge CSPLIT partials per row, compute loss, block-reduce to the scalar */
__global__ __launch_bounds__(1024) void k_loss_final(
    const float* __restrict__ part, const int* __restrict__ rowidx,
    const float* __restrict__ cntb, const float* __restrict__ gcnt,
    float ca, float cg, float c1, float* __restrict__ out_slot) {
  __shared__ float red[1024];
  int row = threadIdx.x;
  int rt = row >> 4, r = row & 15;
  float M = -1e30f, S = 0.0f, P = 0.0f;
  for (int p = 0; p < CSPLIT; ++p) {
    int idx = (p * NRT + rt) * 16 + r;
    float mw = part[idx];
    float mn = fmaxf(M, mw);
    S = S * __expf(M - mn) + part[PART_PLANE + idx] * __expf(mw - mn);
    M = mn;
    P += part[2 * PART_PLANE + idx];
  }
  float lse = M + __logf(S);
  int ro = rowidx[row];
  float np = ca * cntb[ro] + cg * gcnt[ro] + c1;
  red[row] = lse - P / np;
  __syncthreads();
  for (int s = 512; s > 0; s >>= 1) { if (row < s) red[row] += red[row + s]; __syncthreads(); }
  if (row == 0) *out_slot = red[0] * (1.0f / (float)BROWS);
}

extern "C" void kernel_launch(void* const* d_in, const int* in_sizes, int n_in,
                              void* d_out, int out_size, void* d_ws, size_t ws_size,
                              hipStream_t stream) {
  (void)in_sizes; (void)n_in; (void)out_size; (void)ws_size;
  const float* anch = (const float*)d_in[0];
  const float* anchm = (const float*)d_in[1];
  const float* assm = (const float*)d_in[2];
  const float* queue = (const float*)d_in[3];
  const int* bidx = (const int*)d_in[4];
  const int* qidx = (const int*)d_in[5];
  float* out = (float*)d_out;
  float* ws = (float*)d_ws;

  float* anorm = ws + WS_ANORM;
  float* assetsn = ws + WS_ASSETSN;
  float* invq = ws + WS_INVQ;
  float* gsum = ws + WS_GSUM;
  float* gcnt = ws + WS_GCNT;
  float* cntb = ws + WS_CNTB;
  float* suma = ws + WS_SUMA;
  float* sums = ws + WS_SUMS;
  float* bao = ws + WS_BAO;
  float* colmat2 = ws + WS_COLMAT2;
  int* colidx2 = (int*)(ws + WS_COLIDX2);
  float* part0 = ws + WS_PART;
  float* part1 = part0 + PART_SIZE;
  float* part2 = part1 + PART_SIZE;

  k_zero<<<(WS_ZERO_COUNT + 255) / 256, 256, 0, stream>>>(ws + WS_ZERO_BEGIN, WS_ZERO_COUNT, out);
  k_rows<<<BROWS, 128, 0, stream>>>(anch, anchm, assm, bidx, anorm, assetsn, suma, sums, cntb);
  k_queue<<<QCOLS / 256, 256, 0, stream>>>(queue, qidx, gsum, gcnt, invq);
  k_qoe<<<NORGS, 128, 0, stream>>>(gsum, colmat2, colidx2);
  k_orgemb<<<BROWS, 128, 0, stream>>>(gsum, suma, sums, bidx, bao, colmat2, colidx2);

  dim3 lgrid(NRT, CSPLIT);
  /* a_c_loss: cols = assets_n rows (B) then queue columns (Q, per-col scale) */
  k_loss<BROWS, QCOLS, QCOLS, true><<<lgrid, 256, 0, stream>>>(
      anorm, bidx, assetsn, bidx, queue, invq, qidx, part0);
  /* a_o_c_loss: cols = [bao | bpo | qoe] (4096) */
  k_loss<0, N2COLS, N2COLS, false><<<lgrid, 256, 0, stream>>>(
      anorm, bidx, nullptr, nullptr, colmat2, nullptr, colidx2, part1);
  /* o_c_loss: rows = bao, cols = [bpo | qoe] (3072) */
  k_loss<0, N2COLS - BROWS, N2COLS, false><<<lgrid, 256, 0, stream>>>(
      bao, bidx, nullptr, nullptr, colmat2 + BROWS, nullptr, colidx2 + BROWS, part2);

  k_loss_final<<<1, 1024, 0, stream>>>(part0, bidx, cntb, gcnt, 1.0f, 1.0f, 0.0f, out + 0);
  k_loss_final<<<1, 1024, 0, stream>>>(part1, bidx, cntb, gcnt, 2.0f, 0.0f, 1.0f, out + 1);
  k_loss_final<<<1, 1024, 0, stream>>>(part2, bidx, cntb, gcnt, 1.0f, 0.0f, 1.0f, out + 2);
}